// SGC_7318624272617
// MI455X (gfx1250) — compile-verified
//
#include <hip/hip_runtime.h>

#define KDIM 256
#define CH   64
#define ROWS_PER_BLOCK 32
#define XROW_PAD 260          // 256 + 4 floats pad: conflict-free LDS A-frag reads

typedef float v2f __attribute__((ext_vector_type(2)));
typedef float v8f __attribute__((ext_vector_type(8)));
typedef int   v4i __attribute__((ext_vector_type(4)));

#if defined(__gfx1250__) && __has_builtin(__builtin_amdgcn_global_load_async_to_lds_b128)
#define USE_ASYNC_LDS 1
typedef __attribute__((address_space(1))) v4i* gptr_v4i;   // global int4*
typedef __attribute__((address_space(3))) v4i* lptr_v4i;   // LDS int4*
#else
#define USE_ASYNC_LDS 0
#endif

// ---------------- degree / norm ----------------

__global__ void __launch_bounds__(256)
deg_init_kernel(float* __restrict__ deg, int n) {
    int i = blockIdx.x * blockDim.x + threadIdx.x;
    if (i < n) deg[i] = 1.0f;                       // self-loop contributes 1
}

__global__ void __launch_bounds__(256)
deg_count_kernel(const long long* __restrict__ col, float* __restrict__ deg, int E) {
    int e = blockIdx.x * blockDim.x + threadIdx.x;
    if (e < E) atomicAdd(&deg[(int)col[e]], 1.0f);
}

__global__ void __launch_bounds__(256)
dinv_kernel(float* __restrict__ deg, int n) {
    int i = blockIdx.x * blockDim.x + threadIdx.x;
    if (i < n) deg[i] = rsqrtf(deg[i]);             // deg >= 1 always
}

// ---------------- GEMM: h = x @ W + b, fp32 WMMA 16x16x4 ----------------
// Block = 128 threads (4 waves). Block computes 32 rows x 64 cols.
// Wave w owns column tile w (16 cols) and BOTH row tiles -> B-frag loaded
// once, feeds two WMMAs (acc0/acc1 interleave breaks the dep chain).
// x panel (32 rows x 256 = 32KB, contiguous in global) is staged into LDS
// via async global->LDS DMA (ASYNCcnt), padded rows for bank-conflict-free
// A-frag reads.
//
// WMMA f32 16x16x4 layouts:
//  A: lane L<16 holds A[M=L][K=k,k+1]; lane L+16 holds A[M=L][K=k+2,k+3]
//  B: lane L<16 holds B[K=k,k+1][N=L]; lane L+16 also holds B[K=k,k+1][N=L]
//  C/D: vgpr r, lane L<16 -> C[M=r][N=L]; lane L+16 -> C[M=8+r][N=L]

__global__ void __launch_bounds__(128)
gemm_wmma_f32(const float* __restrict__ x, const float* __restrict__ w,
              const float* __restrict__ bias, float* __restrict__ h, int nrows) {
    __shared__ float xs[ROWS_PER_BLOCK * XROW_PAD];     // ~33 KB

    const int tid     = threadIdx.x;
    const int wave    = tid >> 5;                       // 0..3 -> column tile
    const int lane    = tid & 31;
    const int rowBase = blockIdx.x * ROWS_PER_BLOCK;
    const int colBase = wave * 16;
    if (rowBase >= nrows) return;                       // uniform per block

    // ---- stage x[rowBase : rowBase+32, :] -> LDS (one contiguous 32KB read)
    const float* gsrc = x + (size_t)rowBase * KDIM;
    const int nF4 = (ROWS_PER_BLOCK * KDIM) / 4;        // 2048 float4 chunks
#if USE_ASYNC_LDS
    float* gsrc_nc = const_cast<float*>(gsrc);
#pragma unroll 4
    for (int j = tid; j < nF4; j += 128) {
        int r    = j >> 6;                              // 64 float4 per row
        int cpos = (j & 63) * 4;
        __builtin_amdgcn_global_load_async_to_lds_b128(
            (gptr_v4i)(gsrc_nc + (size_t)j * 4),
            (lptr_v4i)(xs + r * XROW_PAD + cpos),
            0, 0);
    }
#if __has_builtin(__builtin_amdgcn_s_wait_asynccnt)
    __builtin_amdgcn_s_wait_asynccnt(0);
#else
    asm volatile("s_wait_asynccnt 0" ::: "memory");
#endif
    __syncthreads();
#else
    for (int j = tid; j < nF4; j += 128) {
        int r    = j >> 6;
        int cpos = (j & 63) * 4;
        const float4 v = ((const float4*)gsrc)[j];
        *(float4*)(xs + r * XROW_PAD + cpos) = v;
    }
    __syncthreads();
#endif

    const int l15   = lane & 15;
    const int khalf = (lane >> 4) * 2;                  // 0 or 2

    const float* wcol = w + colBase + l15;              // + k*CH per step
    const float* xa0  = xs + l15 * XROW_PAD + khalf;            // rows 0-15
    const float* xa1  = xs + (l15 + 16) * XROW_PAD + khalf;     // rows 16-31

    v8f acc0 = {};
    v8f acc1 = {};
#pragma unroll 8
    for (int k = 0; k < KDIM; k += 4) {
        v2f bm;
        bm.x = wcol[(size_t)k * CH];
        bm.y = wcol[(size_t)(k + 1) * CH];
        v2f a0, a1;
        a0.x = xa0[k];
        a0.y = xa0[k + 1];
        a1.x = xa1[k];
        a1.y = xa1[k + 1];
        acc0 = __builtin_amdgcn_wmma_f32_16x16x4_f32(
                   false, a0, false, bm, (short)0, acc0, false, false);
        acc1 = __builtin_amdgcn_wmma_f32_16x16x4_f32(
                   false, a1, false, bm, (short)0, acc1, false, false);
    }

    const float bv  = bias[colBase + l15];
    const int   rhi = (lane >> 4) * 8;
#pragma unroll
    for (int i = 0; i < 8; ++i) {
        int r0 = rowBase + rhi + i;                     // row tile 0
        int r1 = r0 + 16;                               // row tile 1
        h[(size_t)r0 * CH + colBase + l15] = acc0[i] + bv;
        h[(size_t)r1 * CH + colBase + l15] = acc1[i] + bv;
    }
}

// ---------------- propagation hop ----------------
// init: hout[i,:] = dinv[i]^2 * hin[i,:]   (self-loop term of A_hat)
__global__ void __launch_bounds__(256)
hop_init_kernel(const float* __restrict__ hin, const float* __restrict__ dinv,
                float* __restrict__ hout, int n) {
    int idx = blockIdx.x * blockDim.x + threadIdx.x;    // over n*CH
    if (idx < n * CH) {
        float d = dinv[idx >> 6];
        hout[idx] = d * d * hin[idx];
    }
}

// scatter: hout[col,:] += dinv[row]*dinv[col] * hin[row,:]
// 64 consecutive threads = one edge (contiguous 256B channel row)
__global__ void __launch_bounds__(256)
hop_scatter_kernel(const long long* __restrict__ row, const long long* __restrict__ col,
                   const float* __restrict__ dinv, const float* __restrict__ hin,
                   float* __restrict__ hout, int E) {
    int t  = blockIdx.x * blockDim.x + threadIdx.x;
    int e  = t >> 6;
    int ch = t & 63;
    if (e < E) {
        int r = (int)row[e];                            // wave-uniform -> 1 request
        int c = (int)col[e];
        float wgt = dinv[r] * dinv[c];
        atomicAdd(&hout[(size_t)c * CH + ch], wgt * hin[(size_t)r * CH + ch]);
    }
}

__global__ void __launch_bounds__(256)
copy_kernel(const float* __restrict__ src, float* __restrict__ dst, int n) {
    int i = blockIdx.x * blockDim.x + threadIdx.x;
    if (i < n) dst[i] = src[i];
}

// ---------------- launch ----------------

extern "C" void kernel_launch(void* const* d_in, const int* in_sizes, int n_in,
                              void* d_out, int out_size, void* d_ws, size_t ws_size,
                              hipStream_t stream) {
    const float*     x    = (const float*)d_in[0];
    const float*     w    = (const float*)d_in[1];
    const float*     bias = (const float*)d_in[2];
    const long long* ei   = (const long long*)d_in[3];

    const int N = in_sizes[0] / KDIM;      // 100000
    const int E = in_sizes[3] / 2;         // 3200000
    const long long* row = ei;             // ei[0,:]
    const long long* col = ei + E;         // ei[1,:]
    float* out = (float*)d_out;

    char*  ws   = (char*)d_ws;
    float* dinv = (float*)ws;                                    // N floats
    size_t off  = (((size_t)N * sizeof(float)) + 4095) & ~(size_t)4095;
    float* hbuf = (float*)(ws + off);                            // N*CH floats (h0, reused as h2)

    const int TB = 256;
    const int nodeBlocks = (N + TB - 1) / TB;
    const int edgeBlocks = (E + TB - 1) / TB;
    const int elemN      = N * CH;
    const int elemBlocks = (elemN + TB - 1) / TB;
    const int scatBlocks = (int)(((long long)E * CH + TB - 1) / TB);

    // 1) degree -> dinv_sqrt (self-loops folded into init)
    deg_init_kernel <<<nodeBlocks, TB, 0, stream>>>(dinv, N);
    deg_count_kernel<<<edgeBlocks, TB, 0, stream>>>(col, dinv, E);
    dinv_kernel     <<<nodeBlocks, TB, 0, stream>>>(dinv, N);

    // 2) h0 = x @ W + b  (fp32 WMMA, async-LDS staged A panel)
    gemm_wmma_f32<<<N / ROWS_PER_BLOCK, 128, 0, stream>>>(x, w, bias, hbuf, N);

    // 3) hop 1: d_out = D^-1/2 (A+I) D^-1/2 h0
    hop_init_kernel   <<<elemBlocks, TB, 0, stream>>>(hbuf, dinv, out, N);
    hop_scatter_kernel<<<scatBlocks, TB, 0, stream>>>(row, col, dinv, hbuf, out, E);

    // 4) hop 2: hbuf = D^-1/2 (A+I) D^-1/2 d_out, then copy back to d_out
    hop_init_kernel   <<<elemBlocks, TB, 0, stream>>>(out, dinv, hbuf, N);
    hop_scatter_kernel<<<scatBlocks, TB, 0, stream>>>(row, col, dinv, out, hbuf, E);
    copy_kernel       <<<elemBlocks, TB, 0, stream>>>(hbuf, out, elemN);
}